// Attention_51608327028778
// MI455X (gfx1250) — compile-verified
//
#include <hip/hip_runtime.h>

// Problem constants (from reference setup_inputs)
#define T_   32
#define N_   16
#define V_   64
#define C_   128
#define MT_  32
#define MV_  64
#define NQ_  2048   // V_*T_  rows of Q per scene
#define NK_  2048   // MV_*MT_ rows of K/V per scene

typedef __attribute__((ext_vector_type(2))) float v2f;
typedef __attribute__((ext_vector_type(8))) float v8f;

// CDNA5 fp32 WMMA: D[16x16] = A[16x4] * B[4x16] + C
// 8-arg pattern: (neg_a, A, neg_b, B, c_mod, C, reuse_a, reuse_b)
__device__ __forceinline__ v8f wmma4(v2f a, v2f b, v8f c) {
  return __builtin_amdgcn_wmma_f32_16x16x4_f32(false, a, false, b, (short)0, c,
                                               false, false);
}

// ---------------------------------------------------------------------------
// Kernel 1: fused permute + QKV projection.
// grid.x = 3 * N_ * (NQ_/16); one wave32 per 16-row output tile.
// A-frag layout (f32 16x4): lanes 0-31 -> M = lane&15, VGPR pair holds
// K = k0 + 2*(lane>>4) + {0,1}  -> contiguous float2 per lane.
// B-frag (Wt): B[k][c] = W[c][k] -> contiguous float2 in W's row c.
// C/D layout: VGPR r -> M = r + 8*(lane>>4), N = lane&15.
// ---------------------------------------------------------------------------
__global__ __launch_bounds__(32)
void qkv_proj_kernel(const float* __restrict__ P, const float* __restrict__ M,
                     const float* __restrict__ Wq, const float* __restrict__ bq,
                     const float* __restrict__ Wk, const float* __restrict__ bk,
                     const float* __restrict__ Wv, const float* __restrict__ bv,
                     float* __restrict__ Qb, float* __restrict__ Kb,
                     float* __restrict__ Vb)
{
  const int lane = threadIdx.x & 31;
  const int l15  = lane & 15;
  const int hi   = lane >> 4;

  int b = blockIdx.x;
  const int mat = b >> 11;          // 0:Q 1:K 2:V   (2048 tiles per matrix)
  b &= 2047;
  const int n    = b >> 7;          // scene
  const int rt   = b & 127;         // 16-row tile index
  const int row0 = rt << 4;

  const float* W  = (mat == 0) ? Wq : (mat == 1) ? Wk : Wv;
  const float* bi = (mat == 0) ? bq : (mat == 1) ? bk : bv;
  float*       dst = (mat == 0) ? Qb : (mat == 1) ? Kb : Vb;

  // per-lane gathered source row (permute folded in)
  const int r = row0 + l15;              // local row in [0,2048)
  const int g = r >> 5, s = r & 31;      // (v,t) for Q / (mv,mt) for K,V
  const float* rowp = (mat == 0)
      ? P + ((size_t)s * (N_ * V_)  + (size_t)n * V_  + g) * C_
      : M + ((size_t)s * (N_ * MV_) + (size_t)n * MV_ + g) * C_;

  // accumulator initialized with the bias (bias depends only on N = lane&15)
  v8f acc[8];
#pragma unroll
  for (int ct = 0; ct < 8; ++ct) {
    const float bb = bi[ct * 16 + l15];
    acc[ct] = (v8f){bb, bb, bb, bb, bb, bb, bb, bb};
  }

#pragma unroll 4
  for (int k0 = 0; k0 < C_; k0 += 4) {
    const int kk = k0 + 2 * hi;
    const v2f a = *(const v2f*)(rowp + kk);
#pragma unroll
    for (int ct = 0; ct < 8; ++ct) {
      const v2f bf = *(const v2f*)(W + (size_t)(ct * 16 + l15) * C_ + kk);
      acc[ct] = wmma4(a, bf, acc[ct]);
    }
  }

  float* o = dst + ((size_t)n * NQ_ + row0) * C_;
#pragma unroll
  for (int ct = 0; ct < 8; ++ct)
#pragma unroll
    for (int rr = 0; rr < 8; ++rr)
      o[(size_t)(rr + 8 * hi) * C_ + ct * 16 + l15] = acc[ct][rr];
}

// ---------------------------------------------------------------------------
// Kernel 2: attention. One wave32 per (scene, 16-row q-tile).
// Each 16-key S tile covers exactly one mv segment (mt = h*16 + lane&15),
// so the segmented masked softmax maps to per-lane online max/sum registers.
// Two passes (stats, then weights + A@V) avoid storing the 16x2048 score row.
// ---------------------------------------------------------------------------
__global__ __launch_bounds__(32)
void attn_kernel(const float* __restrict__ Qb, const float* __restrict__ Kb,
                 const float* __restrict__ Vb,
                 const unsigned char* __restrict__ mask,
                 float* __restrict__ out, float* __restrict__ attpart)
{
  __shared__ float atile[16 * 17];   // padded 16x16 transpose staging

  const int lane = threadIdx.x & 31;
  const int l15  = lane & 15;
  const int hi   = lane >> 4;

  const int n  = blockIdx.x >> 7;
  const int qt = blockIdx.x & 127;
  const int q0 = qt << 4;
  const int v  = q0 >> 5;            // tile lies within a single vehicle v

  const float* Q  = Qb + ((size_t)n * NQ_ + q0) * C_;
  const float* K  = Kb + (size_t)n * NK_ * C_;
  const float* Vp = Vb + (size_t)n * NK_ * C_;
  const unsigned char* mk = mask + (size_t)n * MV_;

  // preload the 16x128 Q tile as 32 A-fragments (stays in VGPRs both passes)
  v2f qf[32];
#pragma unroll
  for (int s = 0; s < 32; ++s)
    qf[s] = *(const v2f*)(Q + (size_t)l15 * C_ + 4 * s + 2 * hi);

  // online softmax stats: (h = mt/16 half, rr = q-row VGPR index)
  float mX[2][8], sX[2][8];
#pragma unroll
  for (int h = 0; h < 2; ++h)
#pragma unroll
    for (int rr = 0; rr < 8; ++rr) { mX[h][rr] = -3.0e38f; sX[h][rr] = 0.f; }

  // ---- pass 1: masked, segmented online softmax statistics ----
  for (int mv = 0; mv < MV_; ++mv) {
    if (!mk[mv]) continue;           // wave-uniform skip of masked segment
#pragma unroll
    for (int h = 0; h < 2; ++h) {
      const float* Kt = K + ((size_t)(mv * 2 + h) * 16) * C_;
      v8f sc = (v8f){0, 0, 0, 0, 0, 0, 0, 0};
#pragma unroll
      for (int ks = 0; ks < 32; ++ks) {
        const v2f bf = *(const v2f*)(Kt + (size_t)l15 * C_ + 4 * ks + 2 * hi);
        sc = wmma4(qf[ks], bf, sc);
      }
#pragma unroll
      for (int rr = 0; rr < 8; ++rr) {
        const float x  = sc[rr];
        const float mo = mX[h][rr];
        const float mn = fmaxf(mo, x);
        sX[h][rr] = sX[h][rr] * __expf(mo - mn) + __expf(x - mn);
        mX[h][rr] = mn;
      }
    }
  }

  // ---- pass 2: weights, attention-mean partials, out += A @ V ----
  v8f oacc[8];
#pragma unroll
  for (int ct = 0; ct < 8; ++ct) oacc[ct] = (v8f){0, 0, 0, 0, 0, 0, 0, 0};

  for (int mv = 0; mv < MV_; ++mv) {
    float tilesum = 0.f;
    if (mk[mv]) {
#pragma unroll
      for (int h = 0; h < 2; ++h) {
        const int kt = mv * 2 + h;
        const float* Kt = K + ((size_t)kt * 16) * C_;
        v8f sc = (v8f){0, 0, 0, 0, 0, 0, 0, 0};
#pragma unroll
        for (int ks = 0; ks < 32; ++ks) {
          const v2f bf = *(const v2f*)(Kt + (size_t)l15 * C_ + 4 * ks + 2 * hi);
          sc = wmma4(qf[ks], bf, sc);
        }
        float aw[8];
        float lsum = 0.f;
#pragma unroll
        for (int rr = 0; rr < 8; ++rr) {
          const float w = __expf(sc[rr] - mX[h][rr]) / sX[h][rr];
          aw[rr] = w;
          lsum += w;
        }
#pragma unroll
        for (int off = 16; off > 0; off >>= 1)
          lsum += __shfl_xor(lsum, off, 32);
        tilesum += lsum;

        // transpose the 16x16 A tile (C/D layout -> A layout) via LDS.
        // Same-wave DS ops are in-order; the compiler inserts s_wait_dscnt.
#pragma unroll
        for (int rr = 0; rr < 8; ++rr)
          atile[(rr + 8 * hi) * 17 + l15] = aw[rr];

        const float* Vt = Vp + ((size_t)kt * 16) * C_;
#pragma unroll
        for (int ks = 0; ks < 4; ++ks) {
          v2f af;
          af.x = atile[l15 * 17 + ks * 4 + 2 * hi];
          af.y = atile[l15 * 17 + ks * 4 + 2 * hi + 1];
#pragma unroll
          for (int ct = 0; ct < 8; ++ct) {
            const float* vp = Vt + (size_t)(ks * 4 + 2 * hi) * C_ + ct * 16 + l15;
            v2f bf;
            bf.x = vp[0];
            bf.y = vp[C_];
            oacc[ct] = wmma4(af, bf, oacc[ct]);
          }
        }
      }
    }
    if (lane == 0)
      attpart[((size_t)n * 128 + qt) * MV_ + mv] = tilesum;  // 0 if masked
  }

  // store output with inverse permute: q = v*T + t  ->  out[t][n*V+v][c]
#pragma unroll
  for (int ct = 0; ct < 8; ++ct)
#pragma unroll
    for (int rr = 0; rr < 8; ++rr) {
      const int q = q0 + rr + 8 * hi;
      const int t = q & 31;
      out[((size_t)t * (N_ * V_) + (size_t)n * V_ + v) * C_ + ct * 16 + l15] =
          oacc[ct][rr];
    }
}

// ---------------------------------------------------------------------------
// Kernel 3: deterministic reduction of attention-mean partials.
// att[n][v][mv] = (part[n][2v][mv] + part[n][2v+1][mv]) / (T*mT)
// ---------------------------------------------------------------------------
__global__ void att_finalize_kernel(const float* __restrict__ attpart,
                                    float* __restrict__ att)
{
  const int i = blockIdx.x * blockDim.x + threadIdx.x;
  if (i >= N_ * V_ * MV_) return;
  const int mv = i & 63;
  const int v  = (i >> 6) & 63;
  const int n  = i >> 12;
  const float a0 = attpart[((size_t)n * 128 + 2 * v + 0) * MV_ + mv];
  const float a1 = attpart[((size_t)n * 128 + 2 * v + 1) * MV_ + mv];
  att[i] = (a0 + a1) * (1.0f / (float)(T_ * MT_));
}

// ---------------------------------------------------------------------------
extern "C" void kernel_launch(void* const* d_in, const int* in_sizes, int n_in,
                              void* d_out, int out_size, void* d_ws,
                              size_t ws_size, hipStream_t stream)
{
  (void)in_sizes; (void)n_in; (void)out_size; (void)ws_size;

  const float* P  = (const float*)d_in[0];
  const float* M  = (const float*)d_in[1];
  const unsigned char* mask = (const unsigned char*)d_in[2];  // np.bool_ bytes
  const float* Wq = (const float*)d_in[3];
  const float* bq = (const float*)d_in[4];
  const float* Wk = (const float*)d_in[5];
  const float* bk = (const float*)d_in[6];
  const float* Wv = (const float*)d_in[7];
  const float* bv = (const float*)d_in[8];

  float* out = (float*)d_out;                              // [T, N*V, C]
  float* att = out + (size_t)T_ * (N_ * V_) * C_;          // [N, V, mV]

  // workspace: Q | K | V | att partials  (~48.5 MB)
  float* Qb = (float*)d_ws;
  float* Kb = Qb + (size_t)N_ * NQ_ * C_;
  float* Vb = Kb + (size_t)N_ * NK_ * C_;
  float* attpart = Vb + (size_t)N_ * NK_ * C_;             // [N,128,mV]

  qkv_proj_kernel<<<3 * N_ * (NQ_ / 16), 32, 0, stream>>>(
      P, M, Wq, bq, Wk, bk, Wv, bv, Qb, Kb, Vb);
  attn_kernel<<<N_ * (NQ_ / 16), 32, 0, stream>>>(
      Qb, Kb, Vb, mask, out, attpart);
  att_finalize_kernel<<<(N_ * V_ * MV_ + 255) / 256, 256, 0, stream>>>(
      attpart, att);
}